// VQEMA_9560597201481
// MI455X (gfx1250) — compile-verified
//
#include <hip/hip_runtime.h>

// ---------------- problem constants ----------------
#define DIMS        64
#define NCODES      1024
#define NROWS       (64 * 4096)          // 262144
#define ROWS_PER_WG 128
#define CHUNK       128                  // codes staged in LDS per pass
#define NCHUNKS     (NCODES / CHUNK)     // 8

typedef float v2f __attribute__((ext_vector_type(2)));
typedef float v8f __attribute__((ext_vector_type(8)));

// workspace layout (floats):
//   [0]                 loss accumulator
//   [64    .. 64+1024)  enorm2  (||e_k||^2)
//   [1088  .. 2112)     counts  (segment-sum of ones)
//   [2112  .. 67648)    dw      (segment-sum of x rows, K x D)
#define WS_LOSS   0
#define WS_ENORM  64
#define WS_COUNTS 1088
#define WS_DW     2112

// ---------------------------------------------------------------------------
// Async global -> LDS copy (CDNA5 path, bypasses VGPRs, tracked by ASYNCcnt).
// VDST operand is a VGPR holding the per-lane LDS byte offset; the low 32 bits
// of a generic pointer to a __shared__ object are exactly that offset.
// ---------------------------------------------------------------------------
__device__ __forceinline__ void async_copy_b128(const void* gptr, void* lptr) {
    unsigned int lds_off = (unsigned int)(unsigned long long)lptr;
    asm volatile("global_load_async_to_lds_b128 %0, %1, off"
                 :: "v"(lds_off), "v"(gptr)
                 : "memory");
}

// ---------------------------------------------------------------------------
// Kernel 1: zero scratch + precompute per-code squared norms
// ---------------------------------------------------------------------------
__global__ __launch_bounds__(256) void vq_prep(const float* __restrict__ emb,
                                               float* __restrict__ ws) {
    int t = blockIdx.x * blockDim.x + threadIdx.x;    // 0..1023
    if (t >= NCODES) return;
    ws[WS_COUNTS + t] = 0.0f;
    const float4* e4 = (const float4*)(emb + (size_t)t * DIMS);
    float s = 0.0f;
#pragma unroll
    for (int j = 0; j < DIMS / 4; ++j) {
        float4 v = e4[j];
        s += v.x * v.x + v.y * v.y + v.z * v.z + v.w * v.w;
    }
    ws[WS_ENORM + t] = s;
    float* dw = ws + WS_DW + (size_t)t * DIMS;
#pragma unroll
    for (int d = 0; d < DIMS; ++d) dw[d] = 0.0f;
    if (t == 0) ws[WS_LOSS] = 0.0f;
}

// ---------------------------------------------------------------------------
// Kernel 2: WMMA distance GEMM + argmin + quantized gather + loss/counts/dw
//   8 waves per WG, each wave owns a 16-row tile of x.
//   f32 WMMA 16x16x4: lane = M (A) / N (B), half-wave selects K pair
//   ({0,1} for lanes 0-15, {2,3} for lanes 16-31).
//   Embedding chunks double-buffered in LDS via async global->LDS loads.
// ---------------------------------------------------------------------------
__global__ __launch_bounds__(256) void vq_main(const float* __restrict__ x,
                                               const float* __restrict__ emb,
                                               float* __restrict__ ws,
                                               float* __restrict__ out_q) {
    __shared__ float semb[2][CHUNK * DIMS];   // 2 x 32 KB double buffer
    __shared__ int   sbest[ROWS_PER_WG];

    const float* enorm2 = ws + WS_ENORM;
    float* counts       = ws + WS_COUNTS;
    float* dw           = ws + WS_DW;
    float* loss_acc     = ws + WS_LOSS;

    const int tid    = threadIdx.x;
    const int wave   = tid >> 5;
    const int lane   = tid & 31;
    const int l16    = lane & 15;
    const int h      = lane >> 4;                 // half-wave (K pair select)
    const int wgrow0 = blockIdx.x * ROWS_PER_WG;
    const int myrow  = wgrow0 + wave * 16 + l16;  // A-fragment row for this lane

    // --- preload A fragments: x[myrow, 4j + 2h + {0,1}] for j = 0..15 ---
    float2 xa[16];
    {
        const float* xr = x + (size_t)myrow * DIMS + 2 * h;
#pragma unroll
        for (int j = 0; j < 16; ++j)
            xa[j] = *(const float2*)(xr + 4 * j);
    }

    float minv[8];
    int   mini[8];
#pragma unroll
    for (int r = 0; r < 8; ++r) { minv[r] = 3.4e38f; mini[r] = 0; }

    // --- prefetch chunk 0 into buffer 0 (8 async b128 per thread = 32 KB/WG)
    {
        const float4* src = (const float4*)emb;
        float4* dst = (float4*)semb[0];
#pragma unroll
        for (int it = 0; it < 8; ++it) {
            int idx = it * 256 + tid;
            async_copy_b128(src + idx, dst + idx);
        }
    }

    for (int c = 0; c < NCHUNKS; ++c) {
        __syncthreads();   // everyone done reading the buffer we overwrite next
        if (c + 1 < NCHUNKS) {
            const float4* src =
                (const float4*)(emb + (size_t)(c + 1) * CHUNK * DIMS);
            float4* dst = (float4*)semb[(c + 1) & 1];
#pragma unroll
            for (int it = 0; it < 8; ++it) {
                int idx = it * 256 + tid;
                async_copy_b128(src + idx, dst + idx);
            }
            // async loads complete in order: <=8 left => chunk c has landed
            asm volatile("s_wait_asynccnt 0x8" ::: "memory");
        } else {
            asm volatile("s_wait_asynccnt 0x0" ::: "memory");
        }
        __syncthreads();   // chunk c visible to all waves

        const float* sbuf = semb[c & 1];

        // two column tiles per pass: two independent WMMA accumulator chains
        for (int nt = 0; nt < CHUNK / 16; nt += 2) {
            const float* brow0 = sbuf + (nt * 16 + l16) * DIMS + 2 * h;
            const float* brow1 = brow0 + 16 * DIMS;
            v8f acc0 = {};
            v8f acc1 = {};
#pragma unroll
            for (int j = 0; j < 16; ++j) {               // K = 4j .. 4j+3
                float2 b0 = *(const float2*)(brow0 + 4 * j);
                float2 b1 = *(const float2*)(brow1 + 4 * j);
                v2f a;  a[0]  = xa[j].x; a[1]  = xa[j].y;
                v2f vb0; vb0[0] = b0.x;  vb0[1] = b0.y;
                v2f vb1; vb1[0] = b1.x;  vb1[1] = b1.y;
                acc0 = __builtin_amdgcn_wmma_f32_16x16x4_f32(
                    false, a, false, vb0, (short)0, acc0, false, false);
                acc1 = __builtin_amdgcn_wmma_f32_16x16x4_f32(
                    false, a, false, vb1, (short)0, acc1, false, false);
            }
            const int g0 = c * CHUNK + nt * 16 + l16;
            const float en0 = enorm2[g0];
            const float en1 = enorm2[g0 + 16];
#pragma unroll
            for (int r = 0; r < 8; ++r) {                // ||e||^2 - 2 x.e
                float d0 = en0 - 2.0f * acc0[r];
                if (d0 < minv[r]) { minv[r] = d0; mini[r] = g0; }
                float d1 = en1 - 2.0f * acc1[r];
                if (d1 < minv[r]) { minv[r] = d1; mini[r] = g0 + 16; }
            }
        }
    }

    // --- per-row argmin across the 16 lanes of each half-wave ---
#pragma unroll
    for (int r = 0; r < 8; ++r) {
        float v = minv[r];
        int   i = mini[r];
#pragma unroll
        for (int m = 8; m >= 1; m >>= 1) {
            float v2 = __shfl_xor(v, m, 32);
            int   i2 = __shfl_xor(i, m, 32);
            if (v2 < v || (v2 == v && i2 < i)) { v = v2; i = i2; }
        }
        // C layout: vgpr r, lanes 0-15 -> row r ; lanes 16-31 -> row 8+r
        if (l16 == 0) sbest[wave * 16 + 8 * h + r] = i;
    }
    __syncthreads();

    // --- commitment-loss partial: this lane's 32 elements of its row ---
    {
        int best = sbest[wave * 16 + l16];
        const float* er = emb + (size_t)best * DIMS + 2 * h;
        float lp = 0.0f;
#pragma unroll
        for (int j = 0; j < 16; ++j) {
            float2 ev = *(const float2*)(er + 4 * j);
            float dx = xa[j].x - ev.x;
            float dy = xa[j].y - ev.y;
            lp += dx * dx + dy * dy;
        }
#pragma unroll
        for (int m = 16; m >= 1; m >>= 1) lp += __shfl_xor(lp, m, 32);
        if (lane == 0) atomicAdd(loss_acc, lp);
    }

    // --- per-code counts (segment-sum of ones) ---
    if (tid < ROWS_PER_WG) atomicAdd(&counts[sbest[tid]], 1.0f);

    // --- quantized output: 128 rows x 16 float4, gathered from codebook ---
    {
        const float4* emb4 = (const float4*)emb;
        float4* outq4 = (float4*)out_q;
#pragma unroll
        for (int it = 0; it < (ROWS_PER_WG * DIMS / 4) / 256; ++it) {
            int idx = it * 256 + tid;            // 0..2047
            int row = idx >> 4, q = idx & 15;
            int b = sbest[row];
            outq4[(size_t)(wgrow0 + row) * (DIMS / 4) + q] =
                emb4[(size_t)b * (DIMS / 4) + q];
        }
    }

    // --- dw segment-sum: dw[best[row]] += x[row] (spread over 65K addrs) ---
    for (int it = 0; it < (ROWS_PER_WG * DIMS) / 256; ++it) {
        int idx = it * 256 + tid;                // 0..8191
        int row = idx >> 6, d = idx & 63;
        int b = sbest[row];
        atomicAdd(&dw[b * DIMS + d],
                  x[(size_t)(wgrow0 + row) * DIMS + d]);
    }
}

// ---------------------------------------------------------------------------
// Kernel 3: EMA update + Laplace smoothing + outputs (1 WG, 1024 threads)
// ---------------------------------------------------------------------------
__global__ __launch_bounds__(1024) void vq_finalize(
        const float* __restrict__ ema_dw, const float* __restrict__ ema_ch,
        const int* __restrict__ counter, const float* __restrict__ ws,
        float* __restrict__ out_loss, float* __restrict__ out_emb,
        float* __restrict__ out_dwh, float* __restrict__ out_ch) {
    __shared__ float red[32];
    const int t = threadIdx.x;                    // 0..1023 == code id

    const float* counts = ws + WS_COUNTS;
    const float* dw     = ws + WS_DW;

    int step = counter[0] + 1;
    float bias = 1.0f - powf(0.99f, (float)step);

    float ch = ema_ch[t];
    float cn = ch - (ch - counts[t]) * 0.01f;     // EMA cluster size
    out_ch[t] = cn;
    float avg = cn / bias;

    // sum(avg) over all 1024 codes
    float s = avg;
#pragma unroll
    for (int m = 16; m >= 1; m >>= 1) s += __shfl_xor(s, m, 32);
    if ((t & 31) == 0) red[t >> 5] = s;
    __syncthreads();
    if (t < 32) {
        float w = red[t];
#pragma unroll
        for (int m = 16; m >= 1; m >>= 1) w += __shfl_xor(w, m, 32);
        red[t] = w;                                // red[0] = total
    }
    __syncthreads();
    float n = red[0];

    float csize = (avg + 1e-5f) / (n + (float)NCODES * 1e-5f) * n;
    float inv = 1.0f / (csize * bias);             // (ndw/bias)/csize

#pragma unroll
    for (int d = 0; d < DIMS; ++d) {
        int i = t * DIMS + d;
        float e = ema_dw[i];
        float ndw = e - (e - dw[i]) * 0.01f;       // EMA feature sums
        out_dwh[i] = ndw;
        out_emb[i] = ndw * inv;
    }

    if (t == 0)
        out_loss[0] = 0.25f * ws[WS_LOSS] / (float)((size_t)NROWS * DIMS);
}

// ---------------------------------------------------------------------------
extern "C" void kernel_launch(void* const* d_in, const int* in_sizes, int n_in,
                              void* d_out, int out_size, void* d_ws, size_t ws_size,
                              hipStream_t stream) {
    const float* x       = (const float*)d_in[0];
    const float* emb     = (const float*)d_in[1];
    const float* ema_dw  = (const float*)d_in[2];
    const float* ema_ch  = (const float*)d_in[3];
    const int*   counter = (const int*)d_in[4];

    float* ws  = (float*)d_ws;
    float* out = (float*)d_out;
    float* out_q    = out;                                   // N*D
    float* out_loss = out + (size_t)NROWS * DIMS;            // 1
    float* out_emb  = out_loss + 1;                          // K*D
    float* out_dwh  = out_emb + (size_t)NCODES * DIMS;       // K*D
    float* out_ch   = out_dwh + (size_t)NCODES * DIMS;       // K

    vq_prep<<<(NCODES + 255) / 256, 256, 0, stream>>>(emb, ws);
    vq_main<<<NROWS / ROWS_PER_WG, 256, 0, stream>>>(x, emb, ws, out_q);
    vq_finalize<<<1, 1024, 0, stream>>>(ema_dw, ema_ch, counter, ws,
                                        out_loss, out_emb, out_dwh, out_ch);
}